// Selector_56616258896205
// MI455X (gfx1250) — compile-verified
//
#include <hip/hip_runtime.h>
#include <hip/hip_bf16.h>
#include <math.h>

#define N_ENT 20000
#define N_RELC 500
#define DD 200
#define BB 8

typedef __attribute__((ext_vector_type(2))) float v2f;
typedef __attribute__((ext_vector_type(8))) float v8f;

// ---------------------------------------------------------------------------
// Kernel 0: gather per-batch vectors, build RotatE phases and the six padded
// A-matrices (16 x 200, rows 8..15 zero) for the WMMA GEMM phase.
// ---------------------------------------------------------------------------
__global__ void setup_kernel(const int* __restrict__ batch,
                             const float* __restrict__ eemb,
                             const float* __restrict__ remb,
                             const float* __restrict__ ce_re,
                             const float* __restrict__ ce_im,
                             const float* __restrict__ cr_re,
                             const float* __restrict__ cr_im,
                             const float* __restrict__ de,
                             const float* __restrict__ dr,
                             const float* __restrict__ ke,
                             const float* __restrict__ kr,
                             const float* __restrict__ se_h,
                             const float* __restrict__ se_t,
                             const float* __restrict__ sr,
                             const float* __restrict__ sr_inv,
                             float* __restrict__ pre,
                             float* __restrict__ pim,
                             float* __restrict__ Amat) {
  for (int idx = threadIdx.x; idx < BB * DD; idx += blockDim.x) {
    int b = idx / DD, d = idx % DD;
    int h = batch[b * 48 + 0];   // batch[b][0][0]
    int r = batch[b * 48 + 2];   // batch[b][0][2]
    // RotatE phase rotation
    float hre = eemb[(size_t)h * 2 * DD + d];
    float him = eemb[(size_t)h * 2 * DD + DD + d];
    float ph = remb[r * DD + d];
    float c = cosf(ph), s = sinf(ph);
    pre[idx] = hre * c - him * s;
    pim[idx] = hre * s + him * c;
    // ComplEx s_re / s_im
    float cer = ce_re[(size_t)h * DD + d], cei = ce_im[(size_t)h * DD + d];
    float crr = cr_re[r * DD + d], cri = cr_im[r * DD + d];
    Amat[0 * 16 * DD + b * DD + d] = cer * crr - cei * cri;
    Amat[1 * 16 * DD + b * DD + d] = cer * cri + cei * crr;
    Amat[2 * 16 * DD + b * DD + d] = ke[(size_t)h * DD + d] * kr[r * DD + d];
    Amat[3 * 16 * DD + b * DD + d] = de[(size_t)h * DD + d] * dr[r * DD + d];
    Amat[4 * 16 * DD + b * DD + d] = se_h[(size_t)h * DD + d] * sr[r * DD + d];
    Amat[5 * 16 * DD + b * DD + d] = se_t[(size_t)h * DD + d] * sr_inv[r * DD + d];
  }
  // zero padding rows 8..15 of every A matrix (ws is poisoned with 0xAA)
  for (int idx = threadIdx.x; idx < BB * DD; idx += blockDim.x) {
    int b = 8 + idx / DD, d = idx % DD;
    for (int m = 0; m < 6; ++m) Amat[m * 16 * DD + b * DD + d] = 0.0f;
  }
}

// ---------------------------------------------------------------------------
// Kernel 1: RotatE score, one thread per tail entity, 8 accumulators per
// thread; p_re/p_im (8x200 each) staged in LDS. eemb read exactly once.
// ---------------------------------------------------------------------------
__global__ void rotate_kernel(const float* __restrict__ eemb,
                              const float* __restrict__ pre,
                              const float* __restrict__ pim,
                              float* __restrict__ scores) {
  __shared__ float spre[BB * DD];
  __shared__ float spim[BB * DD];
  for (int i = threadIdx.x; i < BB * DD; i += blockDim.x) {
    spre[i] = pre[i];
    spim[i] = pim[i];
  }
  __syncthreads();
  int t = blockIdx.x * blockDim.x + threadIdx.x;
  if (t >= N_ENT) return;
  float acc[BB];
#pragma unroll
  for (int b = 0; b < BB; ++b) acc[b] = 0.0f;
  const float* row = eemb + (size_t)t * 2 * DD;
  for (int d = 0; d < DD; d += 4) {
    float4 tr = *(const float4*)(row + d);
    float4 ti = *(const float4*)(row + DD + d);
#pragma unroll
    for (int j = 0; j < 4; ++j) {
      float trj = (&tr.x)[j];
      float tij = (&ti.x)[j];
#pragma unroll
      for (int b = 0; b < BB; ++b) {
        float dre = spre[b * DD + d + j] - trj;
        float dim = spim[b * DD + d + j] - tij;
        acc[b] += sqrtf(dre * dre + dim * dim);
      }
    }
  }
#pragma unroll
  for (int b = 0; b < BB; ++b)
    scores[(size_t)(0 * BB + b) * N_ENT + t] = -acc[b];
}

// ---------------------------------------------------------------------------
// Kernel 2: the four GEMM-based scores via V_WMMA_F32_16X16X4_F32.
// One wave (32 threads) per 16x16 output tile. blockIdx.y selects the score.
// A: 16x4 f32 frag -> lane=(k>=2?16:0)+m, vgpr=k%2. B: 4x16 symmetric.
// C/D: vgpr j = row M=j for lanes 0..15 (batch rows 0..7 used).
// ---------------------------------------------------------------------------
__global__ __launch_bounds__(32) void gemm_kernel(const float* __restrict__ Amat,
                                                  const float* __restrict__ ce_re,
                                                  const float* __restrict__ ce_im,
                                                  const float* __restrict__ ke,
                                                  const float* __restrict__ de,
                                                  const float* __restrict__ se_h,
                                                  const float* __restrict__ se_t,
                                                  float* __restrict__ scores) {
  int st = blockIdx.y;       // 0..3
  int t0 = blockIdx.x * 16;  // 20000 % 16 == 0, no masking needed (EXEC all-1)
  const float *B0, *B1 = nullptr;
  int a0, a1 = 0, sid;
  float scale = 1.0f;
  switch (st) {
    case 0:  sid = 1; a0 = 0; B0 = ce_re; a1 = 1; B1 = ce_im; break;
    case 1:  sid = 2; a0 = 2; B0 = ke; break;
    case 2:  sid = 3; a0 = 3; B0 = de; break;
    default: sid = 4; a0 = 4; B0 = se_t; a1 = 5; B1 = se_h; scale = 0.5f; break;
  }
  int lane = threadIdx.x;
  int mrow = lane & 15;
  int khalf = (lane >> 4) << 1;  // 0 or 2
  v8f c = {};
  for (int term = 0; term < 2; ++term) {
    const float* Bm = (term == 0) ? B0 : B1;
    if (Bm == nullptr) break;  // wave-uniform
    const float* Ab = Amat + (size_t)((term == 0) ? a0 : a1) * 16 * DD;
    for (int k0 = 0; k0 < DD; k0 += 4) {
      v2f a, bfrag;
      a.x = Ab[mrow * DD + k0 + khalf];
      a.y = Ab[mrow * DD + k0 + khalf + 1];
      const float* br = Bm + (size_t)(t0 + mrow) * DD + k0 + khalf;
      bfrag.x = br[0];
      bfrag.y = br[1];
      c = __builtin_amdgcn_wmma_f32_16x16x4_f32(
          /*neg_a=*/false, a, /*neg_b=*/false, bfrag,
          /*c_mod=*/(short)0, c, /*reuse_a=*/false, /*reuse_b=*/false);
    }
  }
  if (lane < 16) {
#pragma unroll
    for (int j = 0; j < BB; ++j)
      scores[(size_t)(sid * BB + j) * N_ENT + t0 + lane] = c[j] * scale;
  }
}

// ---------------------------------------------------------------------------
// Kernel 3: per-row (score,b) statistics. min/max/sum/sumsq in one pass,
// then derive range, 1-mean_norm, var_norm (ddof=1) using the identity
// mean(s)=(mean(x)-mn)/rng, var(s)=var(x)/rng^2.
// ---------------------------------------------------------------------------
__global__ void stats_kernel(const float* __restrict__ scores,
                             float* __restrict__ stats) {
  __shared__ float smn[256], smx[256];
  __shared__ double ssum[256], ssq[256];
  int row = blockIdx.x;  // 0..39  (score*8 + b)
  const float* p = scores + (size_t)row * N_ENT;
  float mn = 3.4e38f, mx = -3.4e38f;
  double sum = 0.0, sq = 0.0;
  for (int i = threadIdx.x; i < N_ENT; i += blockDim.x) {
    float v = p[i];
    mn = fminf(mn, v);
    mx = fmaxf(mx, v);
    sum += (double)v;
    sq += (double)v * (double)v;
  }
  int tid = threadIdx.x;
  smn[tid] = mn; smx[tid] = mx; ssum[tid] = sum; ssq[tid] = sq;
  __syncthreads();
  for (int s = 128; s > 0; s >>= 1) {
    if (tid < s) {
      smn[tid] = fminf(smn[tid], smn[tid + s]);
      smx[tid] = fmaxf(smx[tid], smx[tid + s]);
      ssum[tid] += ssum[tid + s];
      ssq[tid] += ssq[tid + s];
    }
    __syncthreads();
  }
  if (tid == 0) {
    float mn0 = smn[0];
    float rng = smx[0] - mn0;
    double mean = ssum[0] / (double)N_ENT;
    double var = (ssq[0] - ssum[0] * ssum[0] / (double)N_ENT) / (double)(N_ENT - 1);
    stats[row * 4 + 0] = mn0;
    stats[row * 4 + 1] = rng;
    stats[row * 4 + 2] = 1.0f - (float)((mean - (double)mn0) / (double)rng);
    stats[row * 4 + 3] = (float)(var / ((double)rng * (double)rng));
  }
}

// ---------------------------------------------------------------------------
// Kernel 4: the tiny 5-expert MLP + weight normalization. One block,
// 320 threads = (expert e = tid/64, output neuron o = tid%64).
// ---------------------------------------------------------------------------
__global__ void mlp_kernel(const float* __restrict__ stats,
                           const float* __restrict__ W0, const float* __restrict__ b0,
                           const float* __restrict__ W1, const float* __restrict__ b1,
                           const float* __restrict__ W2, const float* __restrict__ b2,
                           const float* __restrict__ W3, const float* __restrict__ b3,
                           float* __restrict__ weights) {
  __shared__ float mlp_in[BB][10];
  __shared__ float xA[5][BB][64];
  __shared__ float xB[5][BB][64];
  __shared__ float wbuf[5][BB];
  int tid = threadIdx.x;
  if (tid < BB * 10) {
    int b = tid / 10, i = tid % 10;
    int sc = i / 2, which = i % 2;  // feats: [mdiff, var] per score, in order
    mlp_in[b][i] = stats[(sc * BB + b) * 4 + 2 + which];
  }
  __syncthreads();
  int e = tid / 64, o = tid % 64;  // tid < 320 always
  // layer 0 (no relu)
  for (int b = 0; b < BB; ++b) {
    float s = b0[e * 64 + o];
    for (int i = 0; i < 10; ++i) s += mlp_in[b][i] * W0[(e * 10 + i) * 64 + o];
    xA[e][b][o] = s;
  }
  __syncthreads();
  // layer 1 (relu)
  for (int b = 0; b < BB; ++b) {
    float s = b1[e * 64 + o];
    for (int i = 0; i < 64; ++i) s += xA[e][b][i] * W1[(e * 64 + i) * 64 + o];
    xB[e][b][o] = fmaxf(s, 0.0f);
  }
  __syncthreads();
  // layer 2 (relu)
  for (int b = 0; b < BB; ++b) {
    float s = b2[e * 64 + o];
    for (int i = 0; i < 64; ++i) s += xB[e][b][i] * W2[(e * 64 + i) * 64 + o];
    xA[e][b][o] = fmaxf(s, 0.0f);
  }
  __syncthreads();
  // layer 3 (relu, 64 -> 1)
  if (o == 0) {
    for (int b = 0; b < BB; ++b) {
      float s = b3[e];
      for (int i = 0; i < 64; ++i) s += xA[e][b][i] * W3[e * 64 + i];
      wbuf[e][b] = fmaxf(s, 0.0f);
    }
  }
  __syncthreads();
  // per-batch normalize by max |w| over experts; weights layout [b][e]
  if (tid < BB) {
    int b = tid;
    float m = 0.0f;
    for (int ee = 0; ee < 5; ++ee) m = fmaxf(m, fabsf(wbuf[ee][b]));
    for (int ee = 0; ee < 5; ++ee) weights[b * 5 + ee] = wbuf[ee][b] / m;
  }
}

// ---------------------------------------------------------------------------
// Kernel 5: out[b][t] = sum_e w[b][e] * (score_e[b][t]-mn)/rng
// ---------------------------------------------------------------------------
__global__ void combine_kernel(const float* __restrict__ scores,
                               const float* __restrict__ stats,
                               const float* __restrict__ weights,
                               float* __restrict__ out) {
  __shared__ float sst[5 * BB * 4];
  __shared__ float sw[BB * 5];
  for (int i = threadIdx.x; i < 5 * BB * 4; i += blockDim.x) sst[i] = stats[i];
  for (int i = threadIdx.x; i < BB * 5; i += blockDim.x) sw[i] = weights[i];
  __syncthreads();
  int idx = blockIdx.x * blockDim.x + threadIdx.x;
  if (idx >= BB * N_ENT) return;
  int b = idx / N_ENT;
  int t = idx - b * N_ENT;
  float acc = 0.0f;
#pragma unroll
  for (int e = 0; e < 5; ++e) {
    float mn = sst[(e * BB + b) * 4 + 0];
    float rng = sst[(e * BB + b) * 4 + 1];
    float s = (scores[(size_t)(e * BB + b) * N_ENT + t] - mn) / rng;
    acc += sw[b * 5 + e] * s;
  }
  out[idx] = acc;
}

extern "C" void kernel_launch(void* const* d_in, const int* in_sizes, int n_in,
                              void* d_out, int out_size, void* d_ws, size_t ws_size,
                              hipStream_t stream) {
  const int* batch = (const int*)d_in[0];
  const float* eemb = (const float*)d_in[1];
  const float* remb = (const float*)d_in[2];
  const float* ce_re = (const float*)d_in[3];
  const float* ce_im = (const float*)d_in[4];
  const float* cr_re = (const float*)d_in[5];
  const float* cr_im = (const float*)d_in[6];
  const float* de = (const float*)d_in[7];
  const float* dr = (const float*)d_in[8];
  const float* ke = (const float*)d_in[9];
  const float* kr = (const float*)d_in[10];
  const float* se_h = (const float*)d_in[11];
  const float* se_t = (const float*)d_in[12];
  const float* sr = (const float*)d_in[13];
  const float* sr_inv = (const float*)d_in[14];
  const float* W0 = (const float*)d_in[15];
  const float* b0 = (const float*)d_in[16];
  const float* W1 = (const float*)d_in[17];
  const float* b1 = (const float*)d_in[18];
  const float* W2 = (const float*)d_in[19];
  const float* b2 = (const float*)d_in[20];
  const float* W3 = (const float*)d_in[21];
  const float* b3 = (const float*)d_in[22];
  float* out = (float*)d_out;

  // workspace layout (floats)
  float* ws = (float*)d_ws;
  float* pre = ws;                     // 8*200        = 1600
  float* pim = ws + 1600;              // 8*200        = 1600
  float* Amat = ws + 3200;             // 6*16*200     = 19200
  float* scores = ws + 22400;          // 5*8*20000    = 800000
  float* stats = ws + 822400;          // 5*8*4        = 160
  float* weights = ws + 822560;        // 8*5          = 40

  setup_kernel<<<1, 256, 0, stream>>>(batch, eemb, remb, ce_re, ce_im, cr_re,
                                      cr_im, de, dr, ke, kr, se_h, se_t, sr,
                                      sr_inv, pre, pim, Amat);
  rotate_kernel<<<(N_ENT + 255) / 256, 256, 0, stream>>>(eemb, pre, pim, scores);
  gemm_kernel<<<dim3(N_ENT / 16, 4), 32, 0, stream>>>(Amat, ce_re, ce_im, ke, de,
                                                      se_h, se_t, scores);
  stats_kernel<<<5 * BB, 256, 0, stream>>>(scores, stats);
  mlp_kernel<<<1, 320, 0, stream>>>(stats, W0, b0, W1, b1, W2, b2, W3, b3,
                                    weights);
  combine_kernel<<<(BB * N_ENT + 255) / 256, 256, 0, stream>>>(scores, stats,
                                                               weights, out);
}